// yoloLoss_1322849927465
// MI455X (gfx1250) — compile-verified
//
#include <hip/hip_runtime.h>

typedef __attribute__((ext_vector_type(2))) float v2f;
typedef __attribute__((ext_vector_type(8))) float v8f;

#define TPB 256
// per block: 256 cells * 30 floats = 7680 floats per array (30720 B)
// LDS layout: [0,7680) pred, [7680,15360) target  => 61440 B dynamic LDS
// b128 staging chunks: 1920 per array, 3840 total = 15 iterations of 256 threads

__global__ __launch_bounds__(TPB)
void yolo_main(const float* __restrict__ pred, const float* __restrict__ targ,
               float* __restrict__ partials, int nCells)
{
    extern __shared__ float smem[];
    const int tid = threadIdx.x;
    const int blockCell = blockIdx.x * TPB;
    const int remCells0 = nCells - blockCell;
    const int remCells  = remCells0 < TPB ? remCells0 : TPB;
    const int remBytes  = remCells * 120;   // valid bytes per array for this block

    const char* predB = (const char*)pred + (size_t)blockCell * 120;
    const char* targB = (const char*)targ + (size_t)blockCell * 120;

    // ---- async staging: global -> LDS (CDNA5 ASYNCcnt path) ----
    #pragma unroll
    for (int c = 0; c < 15; ++c) {
        int chunk = c * TPB + tid;                 // 0..3839
        bool isPred = chunk < 1920;
        int sub = isPred ? chunk : chunk - 1920;
        if (sub * 16 < remBytes) {
            unsigned ldsOff = (unsigned)chunk * 16u;
            const char* src = (isPred ? predB : targB) + (size_t)sub * 16;
            asm volatile("global_load_async_to_lds_b128 %0, %1, off"
                         :: "v"(ldsOff), "v"(src) : "memory");
        }
    }
    asm volatile("s_wait_asynccnt 0" ::: "memory");
    __syncthreads();

    // ---- per-cell loss ----
    const int g = blockCell + tid;
    const bool valid = g < nCells;
    float conf = 0.0f, reg = 0.0f, cls = 0.0f;

    if (valid) {
        const float* p = smem + tid * 30;
        const float* t = smem + 7680 + tid * 30;
        float P[30], T[30];
        #pragma unroll
        for (int q = 0; q < 30; ++q) { P[q] = p[q]; T[q] = t[q]; }

        const float STEPC = 1.0f / 7.0f;
        const int i = g % 7;
        const int j = (g / 7) % 7;
        const float gx = (float)i, gy = (float)j;

        // target box0 corners
        float tcx = (T[0] + gx) * STEPC, tcy = (T[1] + gy) * STEPC;
        float t0x1 = tcx - T[2] * 0.5f, t0y1 = tcy - T[3] * 0.5f;
        float t0x2 = tcx + T[2] * 0.5f, t0y2 = tcy + T[3] * 0.5f;
        float areaT = T[2] * T[3];

        float iou[2], pcc[2][4];
        #pragma unroll
        for (int bx = 0; bx < 2; ++bx) {
            const int o = bx * 5;
            float cx = (P[o + 0] + gx) * STEPC, cy = (P[o + 1] + gy) * STEPC;
            float x1 = cx - P[o + 2] * 0.5f, y1 = cy - P[o + 3] * 0.5f;
            float x2 = cx + P[o + 2] * 0.5f, y2 = cy + P[o + 3] * 0.5f;
            pcc[bx][0] = x1; pcc[bx][1] = y1; pcc[bx][2] = x2; pcc[bx][3] = y2;
            float lx = fmaxf(x1, t0x1), ly = fmaxf(y1, t0y1);
            float rx = fminf(x2, t0x2), ry = fminf(y2, t0y2);
            float iw = fmaxf(rx - lx, 0.0f), ih = fmaxf(ry - ly, 0.0f);
            float inter = iw * ih;
            float areaP = (x2 - x1) * (y2 - y1);
            iou[bx] = inter / (areaP + areaT - inter);
        }

        const bool sig   = T[9] > 0.0f;             // target box1 conf > 0
        const bool best0 = iou[0] >= iou[1];        // argmax tie -> box 0
        const float iouMax = best0 ? iou[0] : iou[1];

        #pragma unroll
        for (int bx = 0; bx < 2; ++bx) {
            const int o = bx * 5;
            const bool bestk = (bx == 0) ? best0 : !best0;
            const bool obj0k = T[o + 4] > 0.0f;
            const bool objk  = obj0k && (sig ? bestk : true);

            // confidence term (obj + 0.5*noobj folded)
            float pconf = sig ? (bestk ? iouMax : 0.0f) : P[o + 4];
            float d = pconf - T[o + 4];
            float dc = d * d;
            conf += objk ? dc : 0.5f * dc;

            // coordinate term
            float px_ = sig ? pcc[bx][0] : P[o + 0];
            float py_ = sig ? pcc[bx][1] : P[o + 1];
            float pz_ = sig ? pcc[bx][2] : P[o + 2];
            float pw_ = sig ? pcc[bx][3] : P[o + 3];
            float tx_, ty_, tz_, tw_;
            if (bx == 0) {
                tx_ = sig ? t0x1 : T[0]; ty_ = sig ? t0y1 : T[1];
                tz_ = sig ? t0x2 : T[2]; tw_ = sig ? t0y2 : T[3];
            } else {
                tx_ = T[5]; ty_ = T[6]; tz_ = T[7]; tw_ = T[8];
            }
            if (objk) {
                float dx = px_ - tx_, dy = py_ - ty_;
                float sp2 = sqrtf(pz_), sp3 = sqrtf(pw_);
                float st2 = sqrtf(tz_), st3 = sqrtf(tw_);
                float d2 = st2 - sp2, d3 = st3 - sp3;
                reg += dx * dx + dy * dy + d2 * d2 + d3 * d3;
            }
        }

        if (sig) {
            #pragma unroll
            for (int q = 10; q < 30; ++q) {
                float d = P[q] - T[q];
                cls += d * d;
            }
        }
    }

    // ---- wave32 tree reduction ----
    #pragma unroll
    for (int off = 16; off > 0; off >>= 1) {
        conf += __shfl_down(conf, off);
        reg  += __shfl_down(reg,  off);
        cls  += __shfl_down(cls,  off);
    }

    __syncthreads();   // all LDS cell reads done; safe to reuse smem
    const int wave = tid >> 5, lane = tid & 31;
    if (lane == 0) {
        smem[wave * 3 + 0] = conf;
        smem[wave * 3 + 1] = reg;
        smem[wave * 3 + 2] = cls;
    }
    __syncthreads();
    if (tid == 0) {
        float c = 0.0f, r = 0.0f, k = 0.0f;
        #pragma unroll
        for (int w = 0; w < 8; ++w) {
            c += smem[w * 3 + 0];
            r += smem[w * 3 + 1];
            k += smem[w * 3 + 2];
        }
        partials[blockIdx.x * 3 + 0] = c;
        partials[blockIdx.x * 3 + 1] = r;
        partials[blockIdx.x * 3 + 2] = k;
    }
}

// Deterministic final reduction; last 8-wide combine uses V_WMMA_F32_16X16X4_F32
// (A rows 0..2 hold the 3 components, B = ones => D[m][0] = row-sum of A).
__global__ __launch_bounds__(TPB)
void yolo_final(const float* __restrict__ partials, int nBlocks,
                float* __restrict__ out, float invB)
{
    __shared__ float wsum[8][3];
    const int tid = threadIdx.x;
    float c = 0.0f, r = 0.0f, k = 0.0f;
    for (int idx = tid; idx < nBlocks; idx += TPB) {
        c += partials[idx * 3 + 0];
        r += partials[idx * 3 + 1];
        k += partials[idx * 3 + 2];
    }
    #pragma unroll
    for (int off = 16; off > 0; off >>= 1) {
        c += __shfl_down(c, off);
        r += __shfl_down(r, off);
        k += __shfl_down(k, off);
    }
    const int wave = tid >> 5, lane = tid & 31;
    if (lane == 0) { wsum[wave][0] = c; wsum[wave][1] = r; wsum[wave][2] = k; }
    __syncthreads();

    if (tid < 32) {   // wave 0 only: EXEC all ones inside this branch
        const int m  = tid & 15;       // A-matrix row
        const int hi = tid >> 4;       // lanes 16-31 hold K=2,3
        const int k0 = hi * 2, k1 = hi * 2 + 1;
        float a0 = (m < 3) ? (wsum[k0][m] + wsum[k0 + 4][m]) : 0.0f;
        float a1 = (m < 3) ? (wsum[k1][m] + wsum[k1 + 4][m]) : 0.0f;
        v2f A; A[0] = a0; A[1] = a1;
        v2f Bv; Bv[0] = 1.0f; Bv[1] = 1.0f;
        v8f C = {};
        v8f D = __builtin_amdgcn_wmma_f32_16x16x4_f32(
            false, A, false, Bv, (short)0, C, false, false);
        if (tid == 0) {
            out[0] = D[0] * invB;          // conf = (obj + 0.5*noobj)/B
            out[1] = D[1] * 5.0f * invB;   // reg  = 5*(xy+wh)/B
            out[2] = D[2] * invB;          // cls  = cls/B
        }
    }
}

extern "C" void kernel_launch(void* const* d_in, const int* in_sizes, int n_in,
                              void* d_out, int out_size, void* d_ws, size_t ws_size,
                              hipStream_t stream)
{
    const float* pred = (const float*)d_in[0];
    const float* targ = (const float*)d_in[1];
    float* out = (float*)d_out;
    float* partials = (float*)d_ws;

    const int nCells  = in_sizes[0] / 30;          // B*7*7
    const int nBlocks = (nCells + TPB - 1) / TPB;  // 1568 for B=8192
    const float invB  = 1.0f / (float)(nCells / 49);

    const size_t shmem = 15360 * sizeof(float);    // 61440 B
    yolo_main<<<nBlocks, TPB, shmem, stream>>>(pred, targ, partials, nCells);
    yolo_final<<<1, TPB, 0, stream>>>(partials, nBlocks, out, invB);
}